// OurAttention_32667521253496
// MI455X (gfx1250) — compile-verified
//
#include <hip/hip_runtime.h>

typedef float v2f __attribute__((ext_vector_type(2)));
typedef float v4f __attribute__((ext_vector_type(4)));
typedef float v8f __attribute__((ext_vector_type(8)));

#define EMBED   256
#define NCLS    16
#define BATCH   64
#define SEQ     8192
#define GROUPS  64                       // S-splits per batch
#define TOK_PER_GRP   (SEQ / GROUPS)     // 128 tokens
#define TILES_PER_GRP (TOK_PER_GRP / 16) // 8 tiles of 16 tokens
#define WAVES_PER_BLK 8
#define K2_BLOCKS ((BATCH * GROUPS) / WAVES_PER_BLK) // 512
#define REC_STRIDE 32                    // floats per partial record
#define KB 8                             // K4-chunks per pipeline stage

// ---------------------------------------------------------------- kernel 1
// Wp[e] = sum_j W[e][j] * p[j]   (tiny, one block)
__global__ void wp_kernel(const float* __restrict__ W,
                          const float* __restrict__ p,
                          float* __restrict__ wp) {
    const int e = threadIdx.x;           // 256 threads
    const float* row = W + e * EMBED;
    float s = 0.f;
#pragma unroll 8
    for (int j = 0; j < EMBED; ++j) s = fmaf(row[j], p[j], s);
    wp[e] = s;
}

// ---------------------------------------------------------------- kernel 2
// One wave per (batch, 128-token group). Fused logits + online softmax +
// WMMA token-scores + weighted accumulation. Writes (acc[16], m, Z).
//
// LDS holds a packed per-(chunk,half,class) fragment table:
//   pk[((k*2 + hl)*16 + cls)*4] = { nu[4k+klo][cls], nu[4k+klo+1][cls],
//                                   Wp[4k+klo],      Wp[4k+klo+1] }
// so each K4-chunk needs exactly one ds_load_b128 with an immediate offset.
__global__ __launch_bounds__(256, 1)
void pool_kernel(const float* __restrict__ x,
                 const float* __restrict__ mask,
                 const float* __restrict__ nu,
                 const float* __restrict__ wpg,
                 float* __restrict__ recs) {
    __shared__ float lds_pk[64 * 2 * 16 * 4];    // 32 KB
    const int tid = threadIdx.x;
#pragma unroll
    for (int idx = tid; idx < 64 * 2 * 16; idx += 256) {
        const int k = idx >> 5;
        const int h = (idx >> 4) & 1;
        const int c = idx & 15;
        const int r0 = 4 * k + 2 * h;
        v4f q;
        q.x = nu[r0 * NCLS + c];
        q.y = nu[r0 * NCLS + NCLS + c];
        q.z = wpg[r0];
        q.w = wpg[r0 + 1];
        *(v4f*)(lds_pk + idx * 4) = q;
    }
    __syncthreads();

    const int lane = tid & 31;
    const int wave = tid >> 5;
    const int g    = blockIdx.x * WAVES_PER_BLK + wave;  // 0..4095
    const int b    = g >> 6;                 // / GROUPS
    const int grp  = g & (GROUPS - 1);
    const int row  = lane & 15;              // A-matrix M index for this lane
    const int hl   = lane >> 4;              // which half of the wave
    const int klo  = hl << 1;                // K offset within a 4-chunk: 0 or 2
    const int cls  = lane & 15;              // B/D-matrix N (class) index
    const int lanebase = (hl * 16 + cls) * 4;  // float offset into pk per chunk

    float m   = -__builtin_inff();           // running max
    float Z   = 0.f;                         // running sum of exp
    float acc = 0.f;                         // lane-local class partial

    const long tok0 = (long)b * SEQ + (long)grp * TOK_PER_GRP;
    const float* xrow = x + (tok0 + row) * (long)EMBED + klo;   // tile 0

    v2f abuf0[KB], abuf1[KB];
    auto loadA = [&](v2f* buf, const float* base) {
#pragma unroll
        for (int j = 0; j < KB; ++j) buf[j] = *(const v2f*)(base + 4 * j);
    };

    loadA(abuf0, xrow);                      // prime the pipeline (tile 0)

#pragma unroll 1
    for (int t = 0; t < TILES_PER_GRP; ++t) {
        const long tokbase = tok0 + t * 16;

        v8f d = {0.f, 0.f, 0.f, 0.f, 0.f, 0.f, 0.f, 0.f};
        float lp = 0.f;                      // logit partial (half the E dims)

        auto consume = [&](const v2f* buf, int kc) {
            const float* pstage = lds_pk + kc * 128 + lanebase;
#pragma unroll
            for (int j = 0; j < KB; ++j) {
                const v4f q = *(const v4f*)(pstage + j * 128);  // one b128/chunk
                v2f bfr; bfr.x = q.x; bfr.y = q.y;
                d = __builtin_amdgcn_wmma_f32_16x16x4_f32(
                        false, buf[j], false, bfr, (short)0, d, false, false);
                lp = fmaf(buf[j].x, q.z, lp);
                lp = fmaf(buf[j].y, q.w, lp);
            }
        };

        // ---- ping-pong pipeline: 4 x (2 stages of 8 K4-chunks) ----
#pragma unroll 1
        for (int kp = 0; kp < 4; ++kp) {
            loadA(abuf1, xrow + 4 * (kp * 16 + 8));      // chunks +8..+15
            consume(abuf0, kp * 16);
            if (kp < 3)
                loadA(abuf0, xrow + 4 * (kp * 16 + 16)); // chunks +16..+23
            else if (t + 1 < TILES_PER_GRP)
                loadA(abuf0, xrow + 16 * EMBED);         // next tile, chunks 0..7
            consume(abuf1, kp * 16 + 8);
        }

        // full logit for this lane's row (+ attention mask)
        float lrow = lp + __shfl_xor(lp, 16);
        lrow += mask[tokbase + row];

        // tile max over the 16 rows (both halves hold identical copies)
        float tmax = lrow;
        tmax = fmaxf(tmax, __shfl_xor(tmax, 1));
        tmax = fmaxf(tmax, __shfl_xor(tmax, 2));
        tmax = fmaxf(tmax, __shfl_xor(tmax, 4));
        tmax = fmaxf(tmax, __shfl_xor(tmax, 8));

        const float mnew = fmaxf(m, tmax);
        const float s    = __expf(m - mnew);       // rescale old state
        const float w    = __expf(lrow - mnew);    // this row's weight

        float tsum = w;
        tsum += __shfl_xor(tsum, 1);
        tsum += __shfl_xor(tsum, 2);
        tsum += __shfl_xor(tsum, 4);
        tsum += __shfl_xor(tsum, 8);
        Z = fmaf(Z, s, tsum);

        // acc[class] += sum_r w[r] * T[r][class]; lane holds rows hl*8..hl*8+7
        float accadd = 0.f;
        const int rbase = hl << 3;
#pragma unroll
        for (int j = 0; j < 8; ++j) {
            const float wj = __shfl(w, rbase + j); // weight of row rbase+j
            accadd = fmaf(wj, d[j], accadd);
        }
        acc = fmaf(acc, s, accadd);
        m = mnew;

        xrow += 16 * EMBED;                  // advance to next tile
    }

    // combine the two row-halves -> per-class totals, write partial record
    const float accC = acc + __shfl_xor(acc, 16);
    float* rec = recs + (size_t)g * REC_STRIDE;
    if (lane < 16) rec[lane] = accC;
    if (lane == 0) { rec[16] = m; rec[17] = Z; }
}

// ---------------------------------------------------------------- kernel 3
// Merge the 64 group partials per batch with log-sum-exp rescaling.
// One wave per batch; each lane first folds records i and i+32.
__global__ void finalize_kernel(const float* __restrict__ recs,
                                float* __restrict__ out) {
    const int b = blockIdx.x;
    const int i = threadIdx.x;               // 0..31
    const float* r0 = recs + ((size_t)b * GROUPS + i) * REC_STRIDE;
    const float* r1 = r0 + 32 * REC_STRIDE;

    const float m0 = r0[16], Z0 = r0[17];
    const float m1 = r1[16], Z1 = r1[17];
    const float mi = fmaxf(m0, m1);
    const float e0 = __expf(m0 - mi);
    const float e1 = __expf(m1 - mi);
    const float Zi = Z0 * e0 + Z1 * e1;

    float mstar = mi;
    mstar = fmaxf(mstar, __shfl_xor(mstar, 1));
    mstar = fmaxf(mstar, __shfl_xor(mstar, 2));
    mstar = fmaxf(mstar, __shfl_xor(mstar, 4));
    mstar = fmaxf(mstar, __shfl_xor(mstar, 8));
    mstar = fmaxf(mstar, __shfl_xor(mstar, 16));

    const float sc = __expf(mi - mstar);
    float Zs = Zi * sc;
    Zs += __shfl_xor(Zs, 1);
    Zs += __shfl_xor(Zs, 2);
    Zs += __shfl_xor(Zs, 4);
    Zs += __shfl_xor(Zs, 8);
    Zs += __shfl_xor(Zs, 16);
    const float inv = 1.0f / Zs;

#pragma unroll
    for (int c = 0; c < NCLS; ++c) {
        float v = (r0[c] * e0 + r1[c] * e1) * sc;
        v += __shfl_xor(v, 1);
        v += __shfl_xor(v, 2);
        v += __shfl_xor(v, 4);
        v += __shfl_xor(v, 8);
        v += __shfl_xor(v, 16);
        if (i == 0) out[b * NCLS + c] = v * inv;
    }
}

// ---------------------------------------------------------------- launch
extern "C" void kernel_launch(void* const* d_in, const int* in_sizes, int n_in,
                              void* d_out, int out_size, void* d_ws, size_t ws_size,
                              hipStream_t stream) {
    const float* x    = (const float*)d_in[0];  // [64, 8192, 256]
    const float* mask = (const float*)d_in[1];  // [64, 8192]
    const float* W    = (const float*)d_in[2];  // [256, 256]
    const float* p    = (const float*)d_in[3];  // [256, 1]
    const float* nu   = (const float*)d_in[4];  // [256, 16]
    float* out = (float*)d_out;                 // [64, 16]

    float* wp   = (float*)d_ws;                 // 256 floats
    float* recs = wp + EMBED;                   // 4096 * 32 floats

    wp_kernel<<<1, EMBED, 0, stream>>>(W, p, wp);
    pool_kernel<<<K2_BLOCKS, 256, 0, stream>>>(x, mask, nu, wp, recs);
    finalize_kernel<<<BATCH, 32, 0, stream>>>(recs, out);
}